// ClownSelector_58969900974339
// MI455X (gfx1250) — compile-verified
//
#include <hip/hip_runtime.h>
#include <math.h>

// MoE router: L2-normalize -> causal window-3 mean -> GEMM vs prototypes
// -> top-2 of logits -> pairwise-softmax weights. Fused, x read exactly once.
// Matrix op: V_WMMA_F32_16X16X4_F32 (full fp32 to preserve top-k tie behavior).

typedef float v2f __attribute__((ext_vector_type(2)));
typedef float v8f __attribute__((ext_vector_type(8)));

#define B_    8
#define S_    2048
#define D_    1024
#define E_    64
#define TILE  16
#define WIN   3
#define NROWS (TILE + WIN - 1)     // 18 rows: tile + left halo
#define YS    (D_ + 4)             // padded LDS row stride (floats) -> no 64-bank conflicts
#define LSR   (E_ + 4)             // padded logits row stride

__global__ __launch_bounds__(128) void ClownSelector_router_kernel(
    const float* __restrict__ x,      // [B,S,D]
    const float* __restrict__ prot,   // [E,D]
    float* __restrict__ out)          // [B*S*2] modules (as float) ++ [B*S*2] weights
{
  extern __shared__ float smem[];
  float* ynorm  = smem;                 // NROWS * YS  : normalized tokens
  float* rin    = ynorm + NROWS * YS;   // TILE  * YS  : window-averaged router input
  float* logits = rin   + TILE  * YS;   // TILE  * LSR : logits [token][expert]

  const int tid  = threadIdx.x;
  const int lane = tid & 31;
  const int wave = tid >> 5;            // 4 waves

  const int bid = blockIdx.x;
  const int b   = bid / (S_ / TILE);
  const int t0  = (bid % (S_ / TILE)) * TILE;

  // ---- Phase A: load + L2-normalize 18 token rows into LDS (x read once) ----
  for (int r = wave; r < NROWS; r += 4) {
    int s = t0 + r - (WIN - 1);
    if (s < 0) s = 0;                   // replicate first token of this sequence
    const float4* xr = reinterpret_cast<const float4*>(x + ((size_t)b * S_ + s) * D_);
    float4 v[8];
    float sq = 0.f;
#pragma unroll
    for (int j = 0; j < 8; ++j) {
      v[j] = xr[lane + 32 * j];         // coalesced global_load_b128
      sq += v[j].x * v[j].x + v[j].y * v[j].y + v[j].z * v[j].z + v[j].w * v[j].w;
    }
#pragma unroll
    for (int off = 16; off > 0; off >>= 1)
      sq += __shfl_xor(sq, off, 32);    // wave32 reduction
    const float inv = 1.0f / fmaxf(sqrtf(sq), 1e-8f);   // clip(norm, EPS)
    float4* yr = reinterpret_cast<float4*>(ynorm + r * YS);
#pragma unroll
    for (int j = 0; j < 8; ++j) {
      float4 o;
      o.x = v[j].x * inv; o.y = v[j].y * inv; o.z = v[j].z * inv; o.w = v[j].w * inv;
      yr[lane + 32 * j] = o;
    }
  }
  __syncthreads();

  // ---- Phase B: causal moving mean of 3 -> rin[16][1024] in LDS ----
  for (int idx = tid; idx < TILE * (D_ / 4); idx += 128) {
    const int t  = idx >> 8;            // D_/4 == 256 float4 per row
    const int dq = idx & 255;
    const float4 a = reinterpret_cast<const float4*>(ynorm + (t + 0) * YS)[dq];
    const float4 c = reinterpret_cast<const float4*>(ynorm + (t + 1) * YS)[dq];
    const float4 d = reinterpret_cast<const float4*>(ynorm + (t + 2) * YS)[dq];
    const float inv3 = 1.0f / 3.0f;
    float4 o;
    o.x = (a.x + c.x + d.x) * inv3;
    o.y = (a.y + c.y + d.y) * inv3;
    o.z = (a.z + c.z + d.z) * inv3;
    o.w = (a.w + c.w + d.w) * inv3;
    reinterpret_cast<float4*>(rin + t * YS)[dq] = o;
  }
  __syncthreads();

  // ---- Phase C: 16x64 GEMM tile via V_WMMA_F32_16X16X4_F32 (one expert-tile per wave) ----
  {
    const int m  = lane & 15;           // A: M = lane%16 ; B: N = lane%16
    const int kh = (lane >> 4) << 1;    // lanes 0-15 -> K=0,1 ; lanes 16-31 -> K=2,3
    const float* arow = rin  + (size_t)m * YS;               // 8B-aligned (YS even, kh even)
    const float* prow = prot + (size_t)(wave * 16 + m) * D_; // prototypes hot in WGP$/L2
    v8f acc = {0.f, 0.f, 0.f, 0.f, 0.f, 0.f, 0.f, 0.f};
    for (int k = 0; k < D_; k += 4) {
      v2f a  = *reinterpret_cast<const v2f*>(arow + k + kh); // ds_load_b64
      v2f bb = *reinterpret_cast<const v2f*>(prow + k + kh); // global_load_b64
      acc = __builtin_amdgcn_wmma_f32_16x16x4_f32(false, a, false, bb,
                                                  (short)0, acc, false, false);
    }
    // C layout: VGPR i -> (M=i, N=lane) lanes 0-15 ; (M=8+i, N=lane-16) lanes 16-31
    const int mb = (lane < 16) ? 0 : 8;
#pragma unroll
    for (int i = 0; i < 8; ++i)
      logits[(mb + i) * LSR + wave * 16 + m] = acc[i];
  }
  __syncthreads();

  // ---- Phase D: top-2 + pairwise softmax (full softmax denominator cancels) ----
  if (tid < TILE) {
    const float* lrow = logits + tid * LSR;
    float b0 = -INFINITY, b1 = -INFINITY;
    int i0 = 0, i1 = 0;
    for (int e = 0; e < E_; ++e) {      // strict '>' keeps lowest index on ties (lax.top_k)
      const float v = lrow[e];
      if (v > b0)      { b1 = b0; i1 = i0; b0 = v; i0 = e; }
      else if (v > b1) { b1 = v;  i1 = e; }
    }
    const float e1  = __expf(b1 - b0);
    const float den = 1.0f + e1;
    const float w0  = 1.0f / den;
    const float w1  = e1 / den;
    const size_t tok  = (size_t)b * S_ + (t0 + tid);
    const size_t base = (size_t)B_ * S_ * 2;   // weights follow modules in d_out
    out[tok * 2 + 0] = (float)i0;              // modules as float (tuple concat, fp32 out)
    out[tok * 2 + 1] = (float)i1;
    out[base + tok * 2 + 0] = w0;
    out[base + tok * 2 + 1] = w1;
  }
}

extern "C" void kernel_launch(void* const* d_in, const int* in_sizes, int n_in,
                              void* d_out, int out_size, void* d_ws, size_t ws_size,
                              hipStream_t stream) {
  const float* x    = (const float*)d_in[0];   // [8,2048,1024] fp32
  const float* prot = (const float*)d_in[1];   // [64,1024] fp32
  (void)d_in;  // d_in[2] attn_mask unused (all ones; only feeds logging in reference)
  (void)in_sizes; (void)n_in; (void)out_size; (void)d_ws; (void)ws_size;

  float* out = (float*)d_out;
  const size_t shmem = (size_t)(NROWS * YS + TILE * YS + TILE * LSR) * sizeof(float); // 144160 B (<320KB/WGP)
  dim3 grid(B_ * (S_ / TILE));   // 1024 tiles of 16 tokens
  dim3 block(128);               // 4 waves: one 16-expert WMMA tile each
  ClownSelector_router_kernel<<<grid, block, shmem, stream>>>(x, prot, out);
}